// CorrespondenceContrastiveLoss_44787918962826
// MI455X (gfx1250) — compile-verified
//
#include <hip/hip_runtime.h>

typedef __attribute__((ext_vector_type(2))) float v2f;
typedef __attribute__((ext_vector_type(8))) float v8f;

#define DD   256
#define D2   (256 * 256)
#define D3LL (256LL * 256LL * 256LL)
#define NPTS 131072
#define MARGIN_CONST 0.01f
#define NBLK1 1024   // stage-1 blocks: 1024 blk * 8 waves * 16 pts = 131072

// ---------------------------------------------------------------------------
// Stage 1: each wave32 handles 16 points. Channel-difference rows are packed
// into the V_WMMA_F32_16X16X4_F32 A-operand layout and multiplied against
// themselves (A == B register-wise -> D = A * A^T); the Gram diagonal is the
// per-point squared distance summed over channels. Two chained WMMAs cover
// the 8 channels (K = 0..3 and K = 4..7).
// ---------------------------------------------------------------------------
__global__ __launch_bounds__(256) void ccl_stage1(
    const float* __restrict__ F,   // fix_image_feature   [8][256][256][256]
    const float* __restrict__ M,   // moving_image_feature
    const int*   __restrict__ fp,  // fixed_points  [N][3]
    const int*   __restrict__ np_, // negative_points [N][3]
    float* __restrict__ wsum,      // per-block partial sums   [NBLK1]
    float* __restrict__ wcnt)      // per-block partial counts [NBLK1]
{
    const int tid   = threadIdx.x;
    const int lane  = tid & 31;
    const int wave  = tid >> 5;
    const int base  = (blockIdx.x * 8 + wave) * 16;   // first point of this wave
    const int i     = lane & 15;                      // point-in-wave (row M)
    const int p     = base + i;

    // --- coordinates + mask (reference: all six coords must be in [0,256)) ---
    const int x = fp[3 * p + 0], y = fp[3 * p + 1], z = fp[3 * p + 2];
    const int a = np_[3 * p + 0], b = np_[3 * p + 1], c = np_[3 * p + 2];
    const bool mask = ((unsigned)x < 256u) & ((unsigned)y < 256u) & ((unsigned)z < 256u) &
                      ((unsigned)a < 256u) & ((unsigned)b < 256u) & ((unsigned)c < 256u);

    const int xc = min(max(x, 0), 255), yc = min(max(y, 0), 255), zc = min(max(z, 0), 255);
    const int ac = min(max(a, 0), 255), bc = min(max(b, 0), 255), cc = min(max(c, 0), 255);

    const long long s = (long long)xc * D2 + yc * DD + zc;   // spatial offset, fix
    const long long t = (long long)ac * D2 + bc * DD + cc;   // spatial offset, moving

    // Lanes 0-15 feed K = {0,1} (round 1) / {4,5} (round 2);
    // lanes 16-31 feed K = {2,3} / {6,7}. (A-matrix 16x4 f32 VGPR layout.)
    const int c0 = (lane >> 4) * 2;

    // Issue all 8 scattered loads up-front for maximum MLP (each hits a
    // different 64MB channel slab -> independent HBM transactions).
    const float f0 = F[(long long)(c0 + 0) * D3LL + s];
    const float f1 = F[(long long)(c0 + 1) * D3LL + s];
    const float f2 = F[(long long)(c0 + 4) * D3LL + s];
    const float f3 = F[(long long)(c0 + 5) * D3LL + s];
    const float m0 = M[(long long)(c0 + 0) * D3LL + t];
    const float m1 = M[(long long)(c0 + 1) * D3LL + t];
    const float m2 = M[(long long)(c0 + 4) * D3LL + t];
    const float m3 = M[(long long)(c0 + 5) * D3LL + t];

    v2f a1; a1.x = f0 - m0; a1.y = f1 - m1;   // K = c0, c0+1
    v2f a2; a2.x = f2 - m2; a2.y = f3 - m3;   // K = c0+4, c0+5

    // D = A*A^T accumulated over both channel halves (EXEC is all ones here).
    v8f acc = {};
    acc = __builtin_amdgcn_wmma_f32_16x16x4_f32(false, a1, false, a1,
                                                (short)0, acc, false, false);
    acc = __builtin_amdgcn_wmma_f32_16x16x4_f32(false, a2, false, a2,
                                                (short)0, acc, false, false);

    // Diagonal extraction: row r lives at lane r (r<8, VGPR r) and at
    // lane r+16 (r>=8, VGPR r-8). Diag lanes are 0..7 and 24..31, and for
    // both, (lane & 15) == r, so this lane's own mask/point is the right one.
    int  r    = -1;
    if (lane < 8)        r = lane;
    else if (lane >= 24) r = lane - 24;
    const bool diag = (r >= 0);

    float dv = acc[0];
    dv = (r == 1) ? acc[1] : dv;
    dv = (r == 2) ? acc[2] : dv;
    dv = (r == 3) ? acc[3] : dv;
    dv = (r == 4) ? acc[4] : dv;
    dv = (r == 5) ? acc[5] : dv;
    dv = (r == 6) ? acc[6] : dv;
    dv = (r == 7) ? acc[7] : dv;

    const bool live    = diag && mask;
    const float partial = live ? (MARGIN_CONST - dv) : 0.0f;
    const float cntp    = live ? 1.0f : 0.0f;

    // --- deterministic in-block tree reduction over LDS ---
    __shared__ float ssum[256];
    __shared__ float scnt[256];
    ssum[tid] = partial;
    scnt[tid] = cntp;
    __syncthreads();
#pragma unroll
    for (int off = 128; off > 0; off >>= 1) {
        if (tid < off) {
            ssum[tid] += ssum[tid + off];
            scnt[tid] += scnt[tid + off];
        }
        __syncthreads();
    }
    if (tid == 0) {
        wsum[blockIdx.x] = ssum[0];
        wcnt[blockIdx.x] = scnt[0];
    }
}

// ---------------------------------------------------------------------------
// Stage 2: single block reduces the 1024 per-block partials in a fixed order
// and emits the final scalar loss.
// ---------------------------------------------------------------------------
__global__ __launch_bounds__(256) void ccl_stage2(
    const float* __restrict__ wsum, const float* __restrict__ wcnt,
    float* __restrict__ out)
{
    const int tid = threadIdx.x;
    float s = 0.0f, c = 0.0f;
#pragma unroll
    for (int k = 0; k < NBLK1; k += 256) {
        s += wsum[k + tid];
        c += wcnt[k + tid];
    }
    __shared__ float ssum[256];
    __shared__ float scnt[256];
    ssum[tid] = s;
    scnt[tid] = c;
    __syncthreads();
#pragma unroll
    for (int off = 128; off > 0; off >>= 1) {
        if (tid < off) {
            ssum[tid] += ssum[tid + off];
            scnt[tid] += scnt[tid + off];
        }
        __syncthreads();
    }
    if (tid == 0) {
        const float loss = ssum[0] / (2.0f * scnt[0]) * 10000.0f;
        out[0] = loss;
    }
}

extern "C" void kernel_launch(void* const* d_in, const int* in_sizes, int n_in,
                              void* d_out, int out_size, void* d_ws, size_t ws_size,
                              hipStream_t stream) {
    (void)in_sizes; (void)n_in; (void)out_size; (void)ws_size;
    const float* F   = (const float*)d_in[0];  // fix_image_feature
    const float* M   = (const float*)d_in[1];  // moving_image_feature
    const int*   fp  = (const int*)d_in[2];    // fixed_points
    // d_in[3] = positive_points (unused by the reference)
    const int*   np_ = (const int*)d_in[4];    // negative_points

    float* wsum = (float*)d_ws;                // [NBLK1]
    float* wcnt = wsum + NBLK1;                // [NBLK1]
    float* out  = (float*)d_out;

    ccl_stage1<<<NBLK1, 256, 0, stream>>>(F, M, fp, np_, wsum, wcnt);
    ccl_stage2<<<1, 256, 0, stream>>>(wsum, wcnt, out);
}